// PostProcessor_53377853555318
// MI455X (gfx1250) — compile-verified
//
#include <hip/hip_runtime.h>
#include <hip/hip_bf16.h>
#include <stdint.h>

#define Bn 32
#define Nn 1917
#define Cn 91
#define Kn 256
#define NPAD 2048

__device__ __forceinline__ unsigned float_key(float s) {
    unsigned fb = __float_as_uint(s);
    // total-order transform: monotone map float -> uint
    return (fb & 0x80000000u) ? ~fb : (fb | 0x80000000u);
}
__device__ __forceinline__ float key_float(unsigned hi) {
    unsigned fb = (hi & 0x80000000u) ? (hi ^ 0x80000000u) : ~hi;
    return __uint_as_float(fb);
}

// ---------------- box decode: one thread per (b, n) ----------------
__global__ __launch_bounds__(256) void decode_kernel(
        const float* __restrict__ rel, const float* __restrict__ priors,
        const float* __restrict__ wts, float* __restrict__ boxes) {
    int idx = blockIdx.x * 256 + threadIdx.x;
    if (idx >= Bn * Nn) return;
    int n = idx % Nn;
    float4 p = *(const float4*)(priors + (size_t)n * 4);
    float4 r = *(const float4*)(rel + (size_t)idx * 4);
    float w0 = wts[0], w1 = wts[1], w2 = wts[2], w3 = wts[3];
    float yca = (p.x + p.z) * 0.5f;
    float xca = (p.y + p.w) * 0.5f;
    float ha = p.z - p.x;
    float wa = p.w - p.y;
    float ty = r.x / w0, tx = r.y / w1, th = r.z / w2, tw = r.w / w3;
    float yc = ty * ha + yca;
    float xc = tx * wa + xca;
    float h  = expf(th) * ha;
    float wd = expf(tw) * wa;
    float4 o;
    o.x = yc - h * 0.5f;
    o.y = xc - wd * 0.5f;
    o.z = yc + h * 0.5f;
    o.w = xc + wd * 0.5f;
    *(float4*)(boxes + (size_t)idx * 4) = o;
}

// ---------------- one block per (image, class) task ----------------
__global__ __launch_bounds__(256) void nms_kernel(
        const float* __restrict__ scores, const float* __restrict__ boxes,
        float* __restrict__ out) {
    const int task = blockIdx.x;          // 0 .. B*(C-1)-1
    const int b = task / (Cn - 1);
    const int c = task % (Cn - 1) + 1;    // skip background class 0
    const int tid = threadIdx.x;

    __shared__ unsigned long long keys[NPAD];   // 16 KB
    __shared__ float4 cand[Kn];                 // 4 KB (async-to-LDS target)
    __shared__ float  vals[Kn];
    __shared__ float  area[Kn];
    __shared__ unsigned sup[Kn * 8];            // 8 KB suppression bit-matrix
    __shared__ unsigned keepOut[8];

    // ---- 1. build (score, ~index) keys; pad to 2048 with key=0 ----
    const float* sbase = scores + ((size_t)b * Nn) * Cn + c;   // stride Cn
    for (int n = tid; n < NPAD; n += 256) {
        unsigned long long key = 0ull;
        if (n < Nn) {
            if (n + 256 < Nn)
                __builtin_prefetch(sbase + (size_t)(n + 256) * Cn, 0, 0);
            float s = sbase[(size_t)n * Cn];
            key = ((unsigned long long)float_key(s) << 32) |
                  (unsigned long long)(unsigned)(~n);
        }
        keys[n] = key;
    }
    __syncthreads();

    // ---- 2. bitonic sort ascending (2048 u64 keys) ----
    for (int k = 2; k <= NPAD; k <<= 1) {
        for (int j = k >> 1; j > 0; j >>= 1) {
            for (int t = tid; t < NPAD; t += 256) {
                int ixj = t ^ j;
                if (ixj > t) {
                    unsigned long long a = keys[t];
                    unsigned long long bb = keys[ixj];
                    bool up = ((t & k) == 0);
                    if ((a > bb) == up) { keys[t] = bb; keys[ixj] = a; }
                }
            }
            __syncthreads();
        }
    }

    // ---- 3. top-K: gather candidate boxes into LDS via async DMA ----
    {
        unsigned long long key = keys[NPAD - 1 - tid];  // rank tid (descending)
        unsigned idx = ~(unsigned)(key & 0xFFFFFFFFull);
        vals[tid] = key_float((unsigned)(key >> 32));
        const float* gsrc = boxes + ((size_t)b * Nn + idx) * 4;
        unsigned ldsOff = (unsigned)(uintptr_t)(&cand[tid]);   // low 32 bits = LDS offset
        asm volatile("global_load_async_to_lds_b128 %0, %1, off"
                     :: "v"(ldsOff), "v"((unsigned long long)(uintptr_t)gsrc)
                     : "memory");
    }
    asm volatile("s_wait_asynccnt 0" ::: "memory");
    __syncthreads();

    {
        float4 bx = cand[tid];
        area[tid] = fmaxf(bx.z - bx.x, 0.0f) * fmaxf(bx.w - bx.y, 0.0f);
    }
    __syncthreads();

    // ---- 4. suppression bit-matrix: row i = tid, only words w >= i>>5 ----
    {
        const int i = tid;
        float4 bi = cand[i];
        float  ai = area[i];
        int wstart = i >> 5;                  // wave-uniform
        for (int w = 0; w < wstart; ++w) sup[i * 8 + w] = 0u;
        for (int w = wstart; w < 8; ++w) {
            unsigned bits = 0u;
            int jb = w * 32;
            for (int t2 = 0; t2 < 32; ++t2) {
                int j = jb + t2;
                float4 bj = cand[j];          // wave-uniform j -> LDS broadcast
                float ymin = fmaxf(bi.x, bj.x);
                float xmin = fmaxf(bi.y, bj.y);
                float ymax = fminf(bi.z, bj.z);
                float xmax = fminf(bi.w, bj.w);
                float inter = fmaxf(ymax - ymin, 0.0f) * fmaxf(xmax - xmin, 0.0f);
                float iou = inter / (ai + area[j] - inter + 1e-9f);
                if ((j > i) && (iou > 0.6f)) bits |= (1u << t2);
            }
            sup[i * 8 + w] = bits;
        }
    }
    __syncthreads();

    // ---- 5. sequential keep-scan on 8 lanes of wave 0 (no block barriers) ----
    if (tid < 8) {
        unsigned keep = 0u;
        for (int t2 = 0; t2 < 32; ++t2)
            if (vals[tid * 32 + t2] > 0.3f) keep |= (1u << t2);
        for (int i = 0; i < Kn; ++i) {
            unsigned ownerWord = (unsigned)__shfl((int)keep, i >> 5, 32);
            if ((ownerWord >> (i & 31)) & 1u)
                keep &= ~sup[i * 8 + tid];
        }
        keepOut[tid] = keep;
    }
    __syncthreads();

    // ---- 6. write [K,5] output, exact zeros for suppressed slots ----
    {
        unsigned kept = (keepOut[tid >> 5] >> (tid & 31)) & 1u;
        float4 bx = cand[tid];
        float  v  = vals[tid];
        size_t base = (((size_t)b * (Cn - 1) + (c - 1)) * Kn + tid) * 5;
        out[base + 0] = kept ? bx.x : 0.0f;
        out[base + 1] = kept ? bx.y : 0.0f;
        out[base + 2] = kept ? bx.z : 0.0f;
        out[base + 3] = kept ? bx.w : 0.0f;
        out[base + 4] = kept ? v    : 0.0f;
    }
}

extern "C" void kernel_launch(void* const* d_in, const int* in_sizes, int n_in,
                              void* d_out, int out_size, void* d_ws, size_t ws_size,
                              hipStream_t stream) {
    const float* scores         = (const float*)d_in[0];  // [B,N,C]
    const float* box_regression = (const float*)d_in[1];  // [B,N,4]
    const float* priors         = (const float*)d_in[2];  // [N,4]
    const float* coder_weights  = (const float*)d_in[3];  // [4]
    float* out   = (float*)d_out;                         // [B,C-1,K,5]
    float* boxes = (float*)d_ws;                          // [B,N,4] decoded

    int total = Bn * Nn;
    decode_kernel<<<(total + 255) / 256, 256, 0, stream>>>(
        box_regression, priors, coder_weights, boxes);
    nms_kernel<<<Bn * (Cn - 1), 256, 0, stream>>>(scores, boxes, out);
}